// HG_Attention_IB_83580063580989
// MI455X (gfx1250) — compile-verified
//
#include <hip/hip_runtime.h>
#include <hip/hip_bf16.h>

// ---------------------------------------------------------------------------
// HG_Attention_IB on MI455X (gfx1250).
// Bandwidth-bound problem (~55MB traffic, ~1.4 GFLOP) -> keep everything f32,
// do the two big GEMMs (stacked [emb; emb[perm]] @ w1_bot, H1 @ w2) with
// V_WMMA_F32_16X16X4_F32, one 16x16 C-tile per wave32.
// Scratch GEMM outputs are padded to 1744 rows so the epilogue is 8
// unconditional stores (no per-VGPR EXEC juggling), and output widths are
// template constants so all addressing stays 32-bit.
// ---------------------------------------------------------------------------

typedef float v2f __attribute__((ext_vector_type(2)));
typedef float v8f __attribute__((ext_vector_type(8)));

#define NN   871            // nodes
#define NF   512            // feature size
#define TOTE (NN * NF)      // 445952 elements per meta-path
#define NR   (2 * NN)       // 1742 stacked rows (t0 rows then t1 rows)
#define NRP  1744           // padded row count (multiple of 16)
#define XD   13456          // 116*116

// workspace layout (float offsets)
#define OFF_SPART 0                         // 4*64 score partials
#define OFF_COEF  256                       // coef[4] (=1+attn), attnsum at [4]
#define OFF_XPART 512                       // 53*512 xh partials
#define OFF_XH    (OFF_XPART + 53 * 512)    // xh[512]
#define OFF_C1    (OFF_XH + 512)            // c1[512] = b1 + xh@w1_top
#define OFF_H1    (OFF_C1 + 512)            // [1744,512] padded
#define OFF_H2    (OFF_H1 + NRP * 512)      // [1744,256] padded
#define OFF_O     (OFF_H2 + NRP * 256)      // o[1742]

__device__ __forceinline__ v8f wmma_f32_16x16x4(v2f a, v2f b, v8f c) {
  // (neg_a, A, neg_b, B, c_mod, C, reuse_a, reuse_b)
  return __builtin_amdgcn_wmma_f32_16x16x4_f32(false, a, false, b, (short)0, c,
                                               false, false);
}

__device__ __forceinline__ float block_reduce_256(float v, float* sm) {
  const int t = threadIdx.x;
  sm[t] = v;
  __syncthreads();
  for (int s = 128; s > 0; s >>= 1) {
    if (t < s) sm[t] += sm[t + s];
    __syncthreads();
  }
  return sm[0];
}

// K1: per-(p,chunk) partial of sum_{n,f} tanh(W[n]*z + b[n]) * q[n,f]
__global__ void scores_partial_k(const float* __restrict__ z,
                                 const float* __restrict__ W,
                                 const float* __restrict__ bvec,
                                 const float* __restrict__ q,
                                 float* __restrict__ part) {
  __shared__ float sm[256];
  const int p = blockIdx.y, c = blockIdx.x;       // grid (64, 4)
  const int per = (TOTE + 63) / 64;
  const int start = c * per;
  const int end = min(TOTE, start + per);
  float s = 0.f;
  for (int i = start + (int)threadIdx.x; i < end; i += 256) {
    const int n = i >> 9;                          // i / 512
    const float zt = tanhf(W[n] * z[p * TOTE + i] + bvec[n]);
    s += zt * q[i];
  }
  const float tot = block_reduce_256(s, sm);
  if (threadIdx.x == 0) part[p * 64 + c] = tot;
}

// K1b: reduce partials -> scores -> proportions -> softmax -> coef = 1+attn
__global__ void attn_reduce_k(const float* __restrict__ part,
                              float* __restrict__ coef) {
  __shared__ float sm[256];
  const int t = threadIdx.x;
  sm[t] = part[t];
  __syncthreads();
  if (t < 4) {
    float s = 0.f;
    for (int j = 0; j < 64; ++j) s += sm[t * 64 + j];
    sm[t] = s;  // same-wave lockstep: reads above complete before this store
  }
  __syncthreads();
  if (t == 0) {
    float sc[4] = {sm[0], sm[1], sm[2], sm[3]};
    float tot = sc[0] + sc[1] + sc[2] + sc[3];
    float pr[4], mx = -3.4e38f;
    for (int i = 0; i < 4; ++i) { pr[i] = sc[i] / tot; mx = fmaxf(mx, pr[i]); }
    float e[4], se = 0.f;
    for (int i = 0; i < 4; ++i) { e[i] = expf(pr[i] - mx); se += e[i]; }
    float asum = 0.f;
    for (int i = 0; i < 4; ++i) {
      float a = e[i] / se;
      coef[i] = 1.0f + a;
      asum += a;
    }
    coef[4] = asum;
  }
}

// K2: emb[n,f] = sum_p coef[p] * z[p,n,f]  (written to d_out)
__global__ void emb_k(const float* __restrict__ z,
                      const float* __restrict__ coef,
                      float* __restrict__ emb) {
  const int i = blockIdx.x * 256 + threadIdx.x;
  if (i < TOTE) {
    emb[i] = coef[0] * z[i] + coef[1] * z[TOTE + i] +
             coef[2] * z[2 * TOTE + i] + coef[3] * z[3 * TOTE + i];
  }
}

// K3: xh partials over row-chunks of xp_w (the 27.5MB read)
__global__ void xh_partial_k(const float* __restrict__ x,
                             const float* __restrict__ xp_w,
                             float* __restrict__ xpart) {
  const int c = blockIdx.x;                        // 53 chunks of 254 rows
  const int t = threadIdx.x;                       // 256 threads, 2 cols each
  const int d0 = c * 254;
  const int d1 = min(XD, d0 + 254);
  float a0 = 0.f, a1 = 0.f;
  for (int d = d0; d < d1; ++d) {
    const float xv = x[d];
    const float* wr = xp_w + (size_t)d * NF;
    a0 += xv * wr[t];
    a1 += xv * wr[t + 256];
  }
  xpart[c * NF + t] = a0;
  xpart[c * NF + t + 256] = a1;
}

// K3b: xh[j] = xp_b[j] + sum_c xpart[c][j]
__global__ void xh_reduce_k(const float* __restrict__ xpart,
                            const float* __restrict__ xp_b,
                            float* __restrict__ xh) {
  const int j = blockIdx.x * 256 + threadIdx.x;
  float s = xp_b[j];
  for (int c = 0; c < 53; ++c) s += xpart[c * NF + j];
  xh[j] = s;
}

// K3c: c1[j] = b1[j] + sum_f xh[f] * w1[f, j]   (w1 top half, row-major 1024x512)
__global__ void c1_k(const float* __restrict__ xh,
                     const float* __restrict__ w1,
                     const float* __restrict__ b1,
                     float* __restrict__ c1) {
  const int j = blockIdx.x * 256 + threadIdx.x;
  float s = b1[j];
  for (int f = 0; f < NF; ++f) s += xh[f] * w1[f * NF + j];
  c1[j] = s;
}

// K4/K5: Out[r, n] = relu(bias[n] + sum_k A[rowmap(r), k] * Wm[k, n]), K=512.
// One wave -> one 16x16 C tile via V_WMMA_F32_16X16X4_F32. 4 waves/block span
// 64 output columns. EXEC stays all-ones throughout (ISA requirement for
// WMMA); A-loads of the 2 pad rows are clamped, stores go to a padded buffer.
template <bool USE_PERM, int NCOLS>
__global__ void gemm_relu_k512(const float* __restrict__ A,
                               const int* __restrict__ perm,
                               const float* __restrict__ Wm,
                               const float* __restrict__ bias,
                               float* __restrict__ Out) {
  const int wave = threadIdx.x >> 5;
  const int lane = threadIdx.x & 31;
  const int half = lane >> 4;          // 0: K={0,1}, 1: K={2,3} per WMMA step
  const int l = lane & 15;
  const int m0 = blockIdx.x * 16;
  const int n0 = blockIdx.y * 64 + wave * 16;

  int r = m0 + l;                      // A-fragment row (both halves: M = l)
  if (r >= NR) r = NR - 1;             // clamp the 2 pad rows (load stays valid)
  int ar;
  if constexpr (USE_PERM) ar = (r < NN) ? r : perm[r - NN];
  else                    ar = r;
  const float* arow = A + ar * NF;
  const int kOff = 2 * half;

  v8f acc = {};
  #pragma unroll 8
  for (int k = 0; k < NF; k += 4) {
    v2f a = *(const v2f*)(arow + k + kOff);                // A[l][k+kOff..+1]
    v2f bb;
    bb.x = Wm[(k + kOff) * NCOLS + n0 + l];                // B[k+kOff][n0+l]
    bb.y = Wm[(k + kOff + 1) * NCOLS + n0 + l];
    acc = wmma_f32_16x16x4(a, bb, acc);
  }

  const float badd = bias[n0 + l];
  const int outBase = (m0 + 8 * half) * NCOLS + n0 + l;
  #pragma unroll
  for (int v = 0; v < 8; ++v)          // D: VGPR v -> row m0 + v + 8*half
    Out[outBase + v * NCOLS] = fmaxf(acc[v] + badd, 0.f);
}

// K5b: o[r] = b3 + H2[r,:] . w3   (one wave per row)
__global__ void gemv_o_k(const float* __restrict__ H2,
                         const float* __restrict__ w3,
                         const float* __restrict__ b3,
                         float* __restrict__ o) {
  const int wave = threadIdx.x >> 5;
  const int lane = threadIdx.x & 31;
  const int r = blockIdx.x * 4 + wave;
  if (r >= NR) return;
  const float* row = H2 + r * 256;
  float s = 0.f;
  for (int f = lane; f < 256; f += 32) s += row[f] * w3[f];
  for (int off = 16; off > 0; off >>= 1) s += __shfl_down(s, off, 32);
  if (lane == 0) o[r] = s + b3[0];
}

// K6: loss = relu(mean(o0) - log(mean(exp(o1)) + 1e-8)) + 0.1 * sum(attn)
__global__ void final_loss_k(const float* __restrict__ o,
                             const float* __restrict__ coef,
                             float* __restrict__ loss) {
  __shared__ float sm0[256];
  __shared__ float sm1[256];
  const int t = threadIdx.x;
  float s0 = 0.f, s1 = 0.f;
  for (int r = t; r < NN; r += 256) {
    s0 += o[r];
    s1 += expf(o[NN + r]);
  }
  sm0[t] = s0; sm1[t] = s1;
  __syncthreads();
  for (int s = 128; s > 0; s >>= 1) {
    if (t < s) { sm0[t] += sm0[t + s]; sm1[t] += sm1[t + s]; }
    __syncthreads();
  }
  if (t == 0) {
    const float mean0 = sm0[0] / (float)NN;
    const float meanE = sm1[0] / (float)NN;
    const float mi = mean0 - logf(meanE + 1e-8f);
    loss[0] = fmaxf(mi, 0.f) + 0.1f * coef[4];
  }
}

extern "C" void kernel_launch(void* const* d_in, const int* in_sizes, int n_in,
                              void* d_out, int out_size, void* d_ws, size_t ws_size,
                              hipStream_t stream) {
  const float* z    = (const float*)d_in[0];   // [4,871,512]
  const float* x    = (const float*)d_in[1];   // [1,116,116]
  const float* W    = (const float*)d_in[2];   // [871,1]
  const float* bvec = (const float*)d_in[3];   // [871,1]
  const float* q    = (const float*)d_in[4];   // [871,512]
  const float* xp_w = (const float*)d_in[5];   // [13456,512]
  const float* xp_b = (const float*)d_in[6];   // [512]
  const float* w1   = (const float*)d_in[7];   // [1024,512]
  const float* b1   = (const float*)d_in[8];   // [512]
  const float* w2   = (const float*)d_in[9];   // [512,256]
  const float* b2   = (const float*)d_in[10];  // [256]
  const float* w3   = (const float*)d_in[11];  // [256,1]
  const float* b3   = (const float*)d_in[12];  // [1]
  const int*  perm  = (const int*)d_in[13];    // [871]

  float* out  = (float*)d_out;
  float* emb  = out;                 // [871*512]
  float* loss = out + TOTE;          // [1]

  float* ws    = (float*)d_ws;
  float* spart = ws + OFF_SPART;
  float* coefv = ws + OFF_COEF;
  float* xpart = ws + OFF_XPART;
  float* xh    = ws + OFF_XH;
  float* c1    = ws + OFF_C1;
  float* H1    = ws + OFF_H1;
  float* H2    = ws + OFF_H2;
  float* ovec  = ws + OFF_O;

  scores_partial_k<<<dim3(64, 4), 256, 0, stream>>>(z, W, bvec, q, spart);
  attn_reduce_k<<<1, 256, 0, stream>>>(spart, coefv);
  emb_k<<<(TOTE + 255) / 256, 256, 0, stream>>>(z, coefv, emb);
  xh_partial_k<<<53, 256, 0, stream>>>(x, xp_w, xpart);
  xh_reduce_k<<<2, 256, 0, stream>>>(xpart, xp_b, xh);
  c1_k<<<2, 256, 0, stream>>>(xh, w1, b1, c1);
  // GEMM1: [1742(+2 pad),512] = relu(c1 + [emb; emb[perm]] @ w1_bottom)
  gemm_relu_k512<true, 512><<<dim3(NRP / 16, 8), 128, 0, stream>>>(
      emb, perm, w1 + NF * NF, c1, H1);
  // GEMM2: [1742(+2 pad),256] = relu(b2 + H1 @ w2)
  gemm_relu_k512<false, 256><<<dim3(NRP / 16, 4), 128, 0, stream>>>(
      H1, nullptr, w2, b2, H2);
  gemv_o_k<<<(NR + 3) / 4, 128, 0, stream>>>(H2, w3, b3, ovec);
  final_loss_k<<<1, 256, 0, stream>>>(ovec, coefv, loss);
}